// TreeEncoding_41884521070954
// MI455X (gfx1250) — compile-verified
//
#include <hip/hip_runtime.h>
#include <hip/hip_bf16.h>

#define T_LEN   1024
#define B_SIZE  64
#define D_MODEL 512
#define END_TOK 2

typedef __attribute__((ext_vector_type(4))) float v4f;

// ---------------------------------------------------------------------------
// Phase 1: per-example sequential FIFO queue simulation.
// Queue invariant: pushes always append pairs (t,LEFT),(t,RIGHT) at an even
// tail, so slot s has parent parq[s>>1] and lr = s&1.
// Output: pl[b][t] = (parent(t) << 1) | lr(t).  pl[b][0] is never read.
// ---------------------------------------------------------------------------
__global__ void tree_phase1(const int* __restrict__ tokens, int* __restrict__ pl) {
    __shared__ int s_tok[T_LEN];
    __shared__ int s_parq[T_LEN];
    const int b    = blockIdx.x;
    const int lane = threadIdx.x;           // blockDim.x == 32 (one wave)
    const int* tb  = tokens + b * T_LEN;
    for (int i = lane; i < T_LEN; i += 32) s_tok[i] = tb[i];
    __syncthreads();
    if (lane == 0) {
        int* outp = pl + b * T_LEN;
        outp[0]   = 0;
        s_parq[0] = 0;                      // t=0 pushes (0,L),(0,R)
        int head = 0, tail = 2;
        for (int t = 1; t < T_LEN; ++t) {
            int parent = s_parq[head >> 1];
            int lrb    = head & 1;
            outp[t]    = (parent << 1) | lrb;
            if (s_tok[t] != END_TOK) { s_parq[tail >> 1] = t; tail += 2; }
            if (tail - head > 1) head++;    // pop (t>0 always here)
        }
    }
}

// ---------------------------------------------------------------------------
// Phase 2: one wave per row. Stage pl[b][*] into LDS with the CDNA5 async
// global->LDS path, walk the ancestor chain (uniform LDS reads), emit the
// 512-float row as 4 x b128 stores per lane.
// Column for level k: c = 2k + lr(ancestor_k), k < 256.
// Lane layout: column c lives at seg = c>>7, lane = (c>>2)&31, comp = c&3.
// ---------------------------------------------------------------------------
__global__ void tree_phase2(const int* __restrict__ pl,
                            const float* __restrict__ pscal,
                            float* __restrict__ out) {
    __shared__ int s_pl[T_LEN];
    const int blk   = blockIdx.x;          // 512 blocks
    const int b     = blk >> 3;            // example
    const int chunk = blk & 7;             // 128-row chunk within example
    const int tid   = threadIdx.x;         // 0..255
    const int lane  = tid & 31;
    const int wave  = tid >> 5;            // 0..7

    // Async-copy this example's 4KB parent/lr table into LDS (16B per thread).
    {
        unsigned ldsOff = (unsigned)(unsigned long long)(void*)s_pl + (unsigned)tid * 16u;
        unsigned goff   = (unsigned)tid * 16u;
        const int* gbase = pl + b * T_LEN;
        asm volatile("global_load_async_to_lds_b128 %0, %1, %2"
                     :: "v"(ldsOff), "v"(goff), "s"(gbase)
                     : "memory");
        asm volatile("s_wait_asynccnt 0" ::: "memory");
    }
    __syncthreads();

    // Exact p^c for this lane's 16 owned columns (binary exponentiation).
    const float p = pscal[0];
    float pv[16];
#pragma unroll
    for (int i = 0; i < 4; ++i) {
#pragma unroll
        for (int j = 0; j < 4; ++j) {
            int n = i * 128 + lane * 4 + j;
            float r = 1.0f, base = p;
            while (n) { if (n & 1) r *= base; base *= base; n >>= 1; }
            pv[i * 4 + j] = r;
        }
    }

    for (int it = 0; it < 16; ++it) {
        const int t = chunk * 128 + it * 8 + wave;   // row within example
        unsigned m = 0u;                              // ownership mask (16 cols)
        int cur = t;
        if (cur != 0) {
#pragma unroll 1
            for (int k = 0; k < 256; ++k) {
                int e   = s_pl[cur];                  // uniform LDS broadcast
                int c   = (k << 1) | (e & 1);
                if (((c >> 2) & 31) == lane)
                    m |= 1u << (((c >> 7) << 2) | (c & 3));
                cur = e >> 1;
                if (cur == 0) break;                  // root contributes zeros
            }
        }
        float* orow = out + ((size_t)(b * T_LEN + t)) * D_MODEL;
#pragma unroll
        for (int i = 0; i < 4; ++i) {
            v4f val;
            val.x = (m & (1u << (i * 4 + 0))) ? pv[i * 4 + 0] : 0.0f;
            val.y = (m & (1u << (i * 4 + 1))) ? pv[i * 4 + 1] : 0.0f;
            val.z = (m & (1u << (i * 4 + 2))) ? pv[i * 4 + 2] : 0.0f;
            val.w = (m & (1u << (i * 4 + 3))) ? pv[i * 4 + 3] : 0.0f;
            *reinterpret_cast<v4f*>(orow + i * 128 + lane * 4) = val;
        }
    }
}

extern "C" void kernel_launch(void* const* d_in, const int* in_sizes, int n_in,
                              void* d_out, int out_size, void* d_ws, size_t ws_size,
                              hipStream_t stream) {
    const int*   tokens = (const int*)d_in[0];     // [64,1024] int32
    const float* p      = (const float*)d_in[1];   // [1] float32
    int*   pl  = (int*)d_ws;                       // 64*1024 ints = 256 KB scratch
    float* out = (float*)d_out;                    // [64,1024,512] float32

    tree_phase1<<<B_SIZE, 32, 0, stream>>>(tokens, pl);
    tree_phase2<<<B_SIZE * 8, 256, 0, stream>>>(pl, p, out);
}